// BatchedSemiringGraphConvKDDCup_63290638074667
// MI455X (gfx1250) — compile-verified
//
#include <hip/hip_runtime.h>

#define NUM_NODE 50000
#define NUM_EDGE 1600000
#define DIM 64
#define NTILES (NUM_NODE / 16)   // 3125, exact

typedef __attribute__((ext_vector_type(2))) float v2f;
typedef __attribute__((ext_vector_type(4))) float v4f;
typedef __attribute__((ext_vector_type(8))) float v8f;

// ---------------------------------------------------------------------------
// Kernel 1: initialize accumulators. s = boundary, sq = boundary^2, deg = 0.
// ---------------------------------------------------------------------------
__global__ __launch_bounds__(256) void init_acc_kernel(
    const float* __restrict__ boundary,
    float* __restrict__ s_acc, float* __restrict__ sq_acc,
    float* __restrict__ deg) {
  int idx = blockIdx.x * blockDim.x + threadIdx.x;
  if (idx < NUM_NODE * DIM) {
    float bv = boundary[idx];
    s_acc[idx]  = bv;
    sq_acc[idx] = bv * bv;
  }
  if (idx < NUM_NODE) deg[idx] = 0.0f;
}

// ---------------------------------------------------------------------------
// Kernel 2: per-edge message + scatter via f32 global atomics (L2-resident).
// 16 threads per edge, each handles 4 contiguous dims as a float4.
// ---------------------------------------------------------------------------
__global__ __launch_bounds__(256) void edge_scatter_kernel(
    const float* __restrict__ input,
    const int*   __restrict__ edge_list,
    const float* __restrict__ edge_weight,
    const float* __restrict__ relw,
    float* s_acc, float* sq_acc, float* deg) {
  int t = blockIdx.x * blockDim.x + threadIdx.x;
  if (t >= NUM_EDGE * 16) return;
  int e   = t >> 4;
  int sub = t & 15;

  int ni = edge_list[e * 3 + 0];
  int no = edge_list[e * 3 + 1];
  int r  = edge_list[e * 3 + 2];
  float ew = edge_weight[e];

  const v4f* in4 = (const v4f*)input;
  const v4f* rw4 = (const v4f*)relw;
  v4f m = in4[ni * 16 + sub] * rw4[r * 16 + sub];
  m *= ew;

  float* sdst = s_acc  + no * DIM + sub * 4;
  float* qdst = sq_acc + no * DIM + sub * 4;
#pragma unroll
  for (int j = 0; j < 4; ++j) {
    atomicAdd(sdst + j, m[j]);
    atomicAdd(qdst + j, m[j] * m[j]);
  }
  if (sub == 0) atomicAdd(deg + no, 1.0f);
}

// ---------------------------------------------------------------------------
// Kernel 3: fused per-node update + (update @ W^T + b) via f32 WMMA.
// One wave per 16-node tile. A = update tile (16 x 64), B = W^T (64 x 64),
// processed as 16 K-steps of V_WMMA_F32_16X16X4_F32 across 4 N-tiles.
//
// A-fragment layout (16x4 f32): lanes 0-15 -> M=lane, VGPR{0,1}=K{k0,k0+1};
//                               lanes 16-31 -> M=lane-16, K{k0+2,k0+3}.
// B-fragment (4x16, K x N):     lanes 0-15 -> K=k0/k0+1, N=lane;
//                               lanes 16-31 -> K=k0+2/k0+3, N=lane-16.
// C/D (16x16 f32): VGPR r -> M=r (lanes 0-15) / M=r+8 (lanes 16-31), N=lane%16.
// ---------------------------------------------------------------------------
__global__ __launch_bounds__(256) void update_gemm_kernel(
    const float* __restrict__ s_acc,
    const float* __restrict__ sq_acc,
    const float* __restrict__ deg,
    const float* __restrict__ W,
    const float* __restrict__ bias,
    float* __restrict__ out) {
  int wave = (blockIdx.x * blockDim.x + threadIdx.x) >> 5;
  if (wave >= NTILES) return;                 // wave-uniform: EXEC all-1s below

  int lane = threadIdx.x & 31;
  int row  = lane & 15;                       // M (for A/C) or N (for B/C)
  int koff = (lane >> 4) << 1;                // 0 or 2
  int tb   = wave * 16;
  int n    = tb + row;

  float deg1 = deg[n] + 1.0f;
  float inv  = 1.0f / deg1;

  v8f acc0, acc1, acc2, acc3;
  {
    float b0 = bias[ 0 + row];
    float b1 = bias[16 + row];
    float b2 = bias[32 + row];
    float b3 = bias[48 + row];
#pragma unroll
    for (int r2 = 0; r2 < 8; ++r2) {
      acc0[r2] = b0; acc1[r2] = b1; acc2[r2] = b2; acc3[r2] = b3;
    }
  }

  const float* srow = s_acc  + n * DIM;
  const float* qrow = sq_acc + n * DIM;
  const float* w0 = W + ( 0 + row) * DIM;     // B cols of N-tile 0
  const float* w1 = W + (16 + row) * DIM;
  const float* w2 = W + (32 + row) * DIM;
  const float* w3 = W + (48 + row) * DIM;

#pragma unroll
  for (int k0 = 0; k0 < DIM; k0 += 4) {
    int d = k0 + koff;
    v2f sv = *(const v2f*)(srow + d);
    v2f qv = *(const v2f*)(qrow + d);
    v2f a;
    float m0 = sv[0] * inv, m1 = sv[1] * inv;
    a[0] = __builtin_sqrtf(fmaxf(qv[0] * inv - m0 * m0, 1e-6f));
    a[1] = __builtin_sqrtf(fmaxf(qv[1] * inv - m1 * m1, 1e-6f));

    v2f b0 = *(const v2f*)(w0 + d);
    v2f b1 = *(const v2f*)(w1 + d);
    v2f b2 = *(const v2f*)(w2 + d);
    v2f b3 = *(const v2f*)(w3 + d);

    acc0 = __builtin_amdgcn_wmma_f32_16x16x4_f32(false, a, false, b0, (short)0, acc0, false, false);
    acc1 = __builtin_amdgcn_wmma_f32_16x16x4_f32(false, a, false, b1, (short)0, acc1, false, false);
    acc2 = __builtin_amdgcn_wmma_f32_16x16x4_f32(false, a, false, b2, (short)0, acc2, false, false);
    acc3 = __builtin_amdgcn_wmma_f32_16x16x4_f32(false, a, false, b3, (short)0, acc3, false, false);
  }

  int mbase = tb + ((lane >> 4) << 3);        // lanes 16-31 write rows M=r+8
#pragma unroll
  for (int r2 = 0; r2 < 8; ++r2) {
    float* orow = out + (mbase + r2) * DIM + row;
    orow[ 0] = acc0[r2];
    orow[16] = acc1[r2];
    orow[32] = acc2[r2];
    orow[48] = acc3[r2];
  }
}

// ---------------------------------------------------------------------------
extern "C" void kernel_launch(void* const* d_in, const int* in_sizes, int n_in,
                              void* d_out, int out_size, void* d_ws, size_t ws_size,
                              hipStream_t stream) {
  (void)in_sizes; (void)n_in; (void)out_size; (void)ws_size;

  const float* input       = (const float*)d_in[0];
  const float* boundary    = (const float*)d_in[1];
  const int*   edge_list   = (const int*)  d_in[2];
  const float* edge_weight = (const float*)d_in[3];
  const float* relw        = (const float*)d_in[4];
  const float* W           = (const float*)d_in[5];
  const float* bias        = (const float*)d_in[6];
  float* out = (float*)d_out;

  float* s_acc  = (float*)d_ws;                       // NUM_NODE*DIM floats
  float* sq_acc = s_acc  + (size_t)NUM_NODE * DIM;    // NUM_NODE*DIM floats
  float* deg    = sq_acc + (size_t)NUM_NODE * DIM;    // NUM_NODE floats

  {
    int total = NUM_NODE * DIM;
    init_acc_kernel<<<(total + 255) / 256, 256, 0, stream>>>(boundary, s_acc, sq_acc, deg);
  }
  {
    int total = NUM_EDGE * 16;
    edge_scatter_kernel<<<(total + 255) / 256, 256, 0, stream>>>(
        input, edge_list, edge_weight, relw, s_acc, sq_acc, deg);
  }
  {
    int total = NTILES * 32;  // one wave per tile
    update_gemm_kernel<<<(total + 255) / 256, 256, 0, stream>>>(
        s_acc, sq_acc, deg, W, bias, out);
  }
}